// OurDynamics_11751030522781
// MI455X (gfx1250) — compile-verified
//
#include <hip/hip_runtime.h>

typedef __attribute__((ext_vector_type(2))) float v2f;
typedef __attribute__((ext_vector_type(8))) float v8f;

#define T_  4
#define B_  8
#define N_  5000
#define E_  80000
#define NF_ 16
#define L_  3
#define TB_ (T_ * B_)
#define EDGE_TILES_PER_WAVE 4

__device__ __forceinline__ v8f wmma_f32_16x16x4(v2f a, v2f b, v8f c) {
  // 8 args: (neg_a, A, neg_b, B, c_mod, C, reuse_a, reuse_b)
  return __builtin_amdgcn_wmma_f32_16x16x4_f32(false, a, false, b, (short)0, c, false, false);
}

// ds_swizzle_b32 group-of-32: out[lane] = in[lane ^ X]  (and=0x1F, or=0, xor=X)
#define SWZ_XOR(v, X) \
  __int_as_float(__builtin_amdgcn_ds_swizzle(__float_as_int(v), ((X) << 10) | 0x1F))

// h0 = relu(charges * W_embed) replicated per graph; agg = 0; vec = vs
__global__ void init_kernel(const float* __restrict__ vs,
                            const float* __restrict__ charges,
                            const float* __restrict__ W_embed,
                            float* __restrict__ h,
                            float* __restrict__ agg,
                            float* __restrict__ vec) {
  int i = blockIdx.x * blockDim.x + threadIdx.x;
  const int totH = TB_ * N_ * NF_;
  if (i < totH) {
    int g   = i / (N_ * NF_);
    int rem = i - g * (N_ * NF_);
    int n   = rem / NF_;
    int f   = rem - n * NF_;
    int b   = g % B_;
    float v = charges[b * N_ + n] * W_embed[f];
    h[i]   = v > 0.f ? v : 0.f;
    agg[i] = 0.f;
  }
  const int totV = TB_ * N_ * 3;  // totV < totH, same grid covers it
  if (i < totV) vec[i] = vs[i];
}

// Per-edge: m = relu([h_src|h_dst|d2] @ W1 + b1); scatter m into agg[dst],
// scatter d * (m @ Wv) into vec[dst].  One wave = EDGE_TILES_PER_WAVE 16-edge tiles.
__global__ __launch_bounds__(256) void edge_kernel(
    const float* __restrict__ h, float* __restrict__ agg, float* __restrict__ vec,
    const float* __restrict__ xs,
    const int* __restrict__ esrc, const int* __restrict__ edst,
    const float* __restrict__ W1, const float* __restrict__ b1,
    const float* __restrict__ Wv, int layer)
{
  __shared__ float sW1[36][16];      // W1[layer] padded 33 -> 36 rows
  __shared__ float sD[8][16][3];     // per-wave: d vector per edge row
  __shared__ int   sDst[8][16];      // per-wave: dst node (-1 = invalid)

  const int tid = threadIdx.x;
  for (int idx = tid; idx < 36 * 16; idx += 256) {
    int r = idx >> 4, c = idx & 15;
    sW1[r][c] = (r < 33) ? W1[(layer * 33 + r) * 16 + c] : 0.f;
  }
  __syncthreads();

  const int wid  = tid >> 5;
  const int lane = tid & 31;
  const int half = lane >> 4;
  const int row  = lane & 15;   // edge row within tile (A operand)
  const int col  = lane & 15;   // feature column (B/C/D operand)

  const float bias = b1[layer * NF_ + col];
  const float wvc  = Wv[layer * NF_ + col];

  const long tilesPerGraph = E_ / 16;                 // 5000 (exact)
  const long totalTiles    = (long)TB_ * tilesPerGraph;
  const long tile0 = ((long)blockIdx.x * 8 + wid) * EDGE_TILES_PER_WAVE;

  for (int it = 0; it < EDGE_TILES_PER_WAVE; ++it) {
    const long tile   = tile0 + it;
    const bool tvalid = (tile < totalTiles);
    const long tl     = tvalid ? tile : 0;
    const int  g      = (int)(tl / tilesPerGraph);
    const int  ebase  = (int)(tl % tilesPerGraph) * 16;

    const float* hg   = h   + (size_t)g * N_ * NF_;
    const float* xg   = xs  + (size_t)g * N_ * 3;
    float*       aggG = agg + (size_t)g * N_ * NF_;
    float*       vecG = vec + (size_t)g * N_ * 3;

    const int  e      = ebase + row;
    const bool evalid = tvalid && (e < E_);
    const int  ec     = evalid ? e : 0;
    const int  es     = esrc[ec];
    const int  ed     = edst[ec];
    const float dx = xg[ed * 3 + 0] - xg[es * 3 + 0];
    const float dy = xg[ed * 3 + 1] - xg[es * 3 + 1];
    const float dz = xg[ed * 3 + 2] - xg[es * 3 + 2];
    const float d2 = dx * dx + dy * dy + dz * dz;

    if (half == 0) {
      sDst[wid][row]  = evalid ? ed : -1;
      sD[wid][row][0] = dx; sD[wid][row][1] = dy; sD[wid][row][2] = dz;
    }
    __syncthreads();   // uniform: all waves run identical trip count

    const float* hs = hg + (size_t)es * NF_;
    const float* hd = hg + (size_t)ed * NF_;

    // A layout (16x4 f32): lane holds row=lane&15, K = 2*half + {0,1}
    // B layout (4x16 f32): lane holds col=lane&15, K = 2*half + {0,1}
    v8f acc = {};
#pragma unroll
    for (int k = 0; k < 9; ++k) {
      const int c0 = 4 * k + 2 * half;
      float a0, a1;
      if (c0 < 16)       { a0 = hs[c0];      a1 = hs[c0 + 1];  }
      else if (c0 < 32)  { a0 = hd[c0 - 16]; a1 = hd[c0 - 15]; }
      else if (c0 == 32) { a0 = d2;          a1 = 0.f;         }
      else               { a0 = 0.f;         a1 = 0.f;         }
      v2f a, b;
      a.x = a0;               a.y = a1;
      b.x = sW1[c0][col];     b.y = sW1[c0 + 1][col];
      acc = wmma_f32_16x16x4(a, b, acc);
    }

    // D layout: acc[r] = m[row = r + 8*half][col]
    float m[8];
#pragma unroll
    for (int r = 0; r < 8; ++r) {
      float v = acc[r] + bias;
      m[r] = v > 0.f ? v : 0.f;
    }

#pragma unroll
    for (int r = 0; r < 8; ++r) {
      // s = m[row] @ Wv : xor-reduce across the 16 lanes of this half
      float sv = m[r] * wvc;
      sv += SWZ_XOR(sv, 1);
      sv += SWZ_XOR(sv, 2);
      sv += SWZ_XOR(sv, 4);
      sv += SWZ_XOR(sv, 8);
      const int rr = r + 8 * half;
      const int dn = sDst[wid][rr];
      if (dn >= 0) {
        atomicAdd(&aggG[dn * NF_ + col], m[r]);
        if (col < 3) atomicAdd(&vecG[dn * 3 + col], sD[wid][rr][col] * sv);
      }
    }
    __syncthreads();   // protect per-wave staging region before next iteration
  }
}

// Per-node: h += relu(agg @ W2 + b2); reset agg to 0 for the next layer.
__global__ __launch_bounds__(256) void node_kernel(
    float* __restrict__ h, float* __restrict__ agg,
    const float* __restrict__ W2, const float* __restrict__ b2, int layer)
{
  __shared__ float sW2[16][16];
  const int tid = threadIdx.x;
  if (tid < 256) {
    int r = tid >> 4, c = tid & 15;
    sW2[r][c] = W2[(layer * 16 + r) * 16 + c];
  }
  __syncthreads();

  const int wid  = tid >> 5;
  const int lane = tid & 31;
  const int half = lane >> 4;
  const int row  = lane & 15;
  const int col  = lane & 15;

  const long tilesPerGraph = (N_ + 15) / 16;          // 313
  const long tile  = (long)blockIdx.x * 8 + wid;
  const int  g     = (int)(tile / tilesPerGraph);
  const int  nbase = (int)(tile % tilesPerGraph) * 16;
  const bool gvalid = (g < TB_);
  const int  gg = gvalid ? g : 0;

  float* hG   = h   + (size_t)gg * N_ * NF_;
  float* aggG = agg + (size_t)gg * N_ * NF_;

  const int node = nbase + row;
  const int nc   = (node < N_) ? node : (N_ - 1);
  const float* arow = aggG + (size_t)nc * NF_;

  v8f acc = {};
#pragma unroll
  for (int k = 0; k < 4; ++k) {
    const int c0 = 4 * k + 2 * half;
    v2f a, b;
    a.x = arow[c0];       a.y = arow[c0 + 1];
    b.x = sW2[c0][col];   b.y = sW2[c0 + 1][col];
    acc = wmma_f32_16x16x4(a, b, acc);
  }

  const float bias = b2[layer * NF_ + col];
#pragma unroll
  for (int r = 0; r < 8; ++r) {
    const int nrow = nbase + r + 8 * half;
    if (gvalid && nrow < N_) {
      const size_t idx = (size_t)nrow * NF_ + col;
      const float u  = acc[r] + bias;
      const float hv = hG[idx];
      hG[idx]   = hv + (u > 0.f ? u : 0.f);
      aggG[idx] = 0.f;   // ready for next layer's atomics
    }
  }
}

// out = softmax(theta) . (vec + xs) over T
__global__ void final_kernel(const float* __restrict__ vec,
                             const float* __restrict__ xs,
                             const float* __restrict__ theta,
                             float* __restrict__ out)
{
  const int i = blockIdx.x * blockDim.x + threadIdx.x;
  const int BN3 = B_ * N_ * 3;
  if (i >= BN3) return;
  float th[T_];
  float mx = -1e30f;
#pragma unroll
  for (int t = 0; t < T_; ++t) { th[t] = theta[t]; mx = fmaxf(mx, th[t]); }
  float se = 0.f;
#pragma unroll
  for (int t = 0; t < T_; ++t) { th[t] = __expf(th[t] - mx); se += th[t]; }
  float acc = 0.f;
#pragma unroll
  for (int t = 0; t < T_; ++t)
    acc += th[t] * (vec[(size_t)t * BN3 + i] + xs[(size_t)t * BN3 + i]);
  out[i] = acc / se;
}

extern "C" void kernel_launch(void* const* d_in, const int* in_sizes, int n_in,
                              void* d_out, int out_size, void* d_ws, size_t ws_size,
                              hipStream_t stream) {
  const float* xs      = (const float*)d_in[0];
  const float* vs      = (const float*)d_in[1];
  const float* charges = (const float*)d_in[2];
  const int*   esrc    = (const int*)d_in[3];
  const int*   edst    = (const int*)d_in[4];
  const float* theta   = (const float*)d_in[5];
  const float* W_embed = (const float*)d_in[6];
  const float* W1      = (const float*)d_in[7];
  const float* b1      = (const float*)d_in[8];
  const float* W2      = (const float*)d_in[9];
  const float* b2      = (const float*)d_in[10];
  const float* Wv      = (const float*)d_in[11];
  float* out = (float*)d_out;

  float* h   = (float*)d_ws;                         // [TB, N, NF]
  float* agg = h   + (size_t)TB_ * N_ * NF_;         // [TB, N, NF]
  float* vec = agg + (size_t)TB_ * N_ * NF_;         // [TB, N, 3]

  {
    const int tot = TB_ * N_ * NF_;
    init_kernel<<<(tot + 255) / 256, 256, 0, stream>>>(vs, charges, W_embed, h, agg, vec);
  }

  const long totalEdgeTiles = (long)TB_ * (E_ / 16);          // 160000
  const int  edgeBlocks = (int)((totalEdgeTiles + 8 * EDGE_TILES_PER_WAVE - 1) /
                                (8 * EDGE_TILES_PER_WAVE));   // 5000
  const int  nodeBlocks = (int)(((long)TB_ * ((N_ + 15) / 16) + 7) / 8);
  for (int l = 0; l < L_; ++l) {
    edge_kernel<<<edgeBlocks, 256, 0, stream>>>(h, agg, vec, xs, esrc, edst, W1, b1, Wv, l);
    node_kernel<<<nodeBlocks, 256, 0, stream>>>(h, agg, W2, b2, l);
  }

  final_kernel<<<(B_ * N_ * 3 + 255) / 256, 256, 0, stream>>>(vec, xs, theta, out);
}